// GNNRegression3_28174985462644
// MI455X (gfx1250) — compile-verified
//
#include <hip/hip_runtime.h>

#define H     128
#define EPS_  1e-5f

typedef float v2f __attribute__((ext_vector_type(2)));
typedef float v8f __attribute__((ext_vector_type(8)));

// ---------------- Layer 1: scalar-feature edge scatter ----------------
__global__ void edge_scatter1(const float* __restrict__ x,
                              const int* __restrict__ src,
                              const int* __restrict__ dst,
                              float* __restrict__ agg1, int E) {
    int e = blockIdx.x * blockDim.x + threadIdx.x;
    if (e < E) atomicAdd(&agg1[dst[e]], x[src[e]]);
}

// ---------------- Layer 1: fused outer-product + BN + ReLU ----------------
__global__ void layer1_node(const float* __restrict__ x, const float* __restrict__ agg1,
                            const float* __restrict__ Wrel, const float* __restrict__ brel,
                            const float* __restrict__ Wroot,
                            const float* __restrict__ gamma, const float* __restrict__ beta,
                            const float* __restrict__ mean,  const float* __restrict__ var,
                            float* __restrict__ h1, int N) {
    int tid = blockIdx.x * blockDim.x + threadIdx.x;
    if (tid >= N * H) return;
    int i = tid >> 7, c = tid & (H - 1);
    float v  = agg1[i] * Wrel[c] + brel[c] + x[i] * Wroot[c];
    float sc = gamma[c] * rsqrtf(var[c] + EPS_);
    v = (v - mean[c]) * sc + beta[c];
    h1[tid] = fmaxf(v, 0.0f);
}

// ---------------- Layers 2/3: 128-wide edge scatter (32 lanes/edge) ----------------
__global__ void edge_scatterH(const float* __restrict__ hin,
                              const int* __restrict__ src,
                              const int* __restrict__ dst,
                              float* __restrict__ agg, int E) {
    int tid  = blockIdx.x * blockDim.x + threadIdx.x;
    int e    = tid >> 5;          // whole wave shares one edge -> uniform s,d
    int lane = tid & 31;          // 4 channels per lane
    if (e >= E) return;
    int s = src[e], d = dst[e];
    const float4 v = *(const float4*)(hin + (size_t)s * H + lane * 4);
    float* o = agg + (size_t)d * H + lane * 4;
    atomicAdd(o + 0, v.x); atomicAdd(o + 1, v.y);
    atomicAdd(o + 2, v.z); atomicAdd(o + 3, v.w);
}

// ---------------- WMMA f32 GEMM: out = BN(agg@Wrel^T + brel + h@Wroot^T) ----------------
// One block = one 16-row stripe; 8 waves each own a 16x16 output tile.
// agg/h row-block staged to LDS (stride 132 to avoid bank conflicts); `out` may
// alias `agg` (in-place) because all reads of agg happen from LDS after the barrier.
template <bool RELU>
__global__ __launch_bounds__(256) void gemm_bn(
        const float* __restrict__ agg, const float* __restrict__ hin,
        const float* __restrict__ Wrel, const float* __restrict__ brel,
        const float* __restrict__ Wroot,
        const float* __restrict__ gamma, const float* __restrict__ beta,
        const float* __restrict__ mean,  const float* __restrict__ var,
        float* __restrict__ out, int N) {
    __shared__ float sAgg[16 * 132];
    __shared__ float sH[16 * 132];
    const int rowBase = blockIdx.x * 16;

    for (int i = threadIdx.x; i < 16 * H; i += 256) {
        int r = i >> 7, c = i & (H - 1);
        int row = rowBase + r; if (row >= N) row = N - 1;   // clamp (N%16==0 normally)
        sAgg[r * 132 + c] = agg[(size_t)row * H + c];
        sH  [r * 132 + c] = hin[(size_t)row * H + c];
    }
    __syncthreads();

    const int wave    = threadIdx.x >> 5;
    const int lane    = threadIdx.x & 31;
    const int colBase = wave * 16;
    const int m    = lane & 15;            // A: row M / B: col N
    const int koff = (lane >> 4) << 1;     // lanes 16-31 hold K+2,K+3
    const int co   = colBase + m;

    const float sc = gamma[co] * rsqrtf(var[co] + EPS_);
    const float bs = beta[co] + (brel[co] - mean[co]) * sc;

    const float* wrelRow  = Wrel  + (size_t)co * H;   // B[k][n] = W[n][k] -> contiguous per lane
    const float* wrootRow = Wroot + (size_t)co * H;

    v8f c8 = {};
#pragma unroll 8
    for (int k = 0; k < H; k += 4) {
        v2f aA = *(const v2f*)&sAgg[m * 132 + k + koff];
        v2f bR = *(const v2f*)&wrelRow[k + koff];
        c8 = __builtin_amdgcn_wmma_f32_16x16x4_f32(false, aA, false, bR, (short)0, c8, false, false);
        v2f aH = *(const v2f*)&sH[m * 132 + k + koff];
        v2f bT = *(const v2f*)&wrootRow[k + koff];
        c8 = __builtin_amdgcn_wmma_f32_16x16x4_f32(false, aH, false, bT, (short)0, c8, false, false);
    }

    const int rowHalf = (lane >> 4) << 3;  // C/D: lanes 16-31 hold M=r+8
#pragma unroll
    for (int r = 0; r < 8; ++r) {
        int row = rowBase + r + rowHalf;
        float v = c8[r] * sc + bs;
        if (RELU) v = fmaxf(v, 0.0f);
        if (row < N) out[(size_t)row * H + co] = v;
    }
}

// ---------------- Global mean pool (sorted batch -> atomics mostly hit L2 lines) ----------------
__global__ void pool_kernel(const float* __restrict__ h3, const int* __restrict__ batch,
                            float* __restrict__ sums, float* __restrict__ cnts, int N) {
    int tid = blockIdx.x * blockDim.x + threadIdx.x;
    if (tid >= N * H) return;
    int node = tid >> 7, c = tid & (H - 1);
    int g = batch[node];
    atomicAdd(&sums[g * H + c], h3[(size_t)node * H + c]);
    if (c == 0) atomicAdd(&cnts[g], 1.0f);
}

// ---------------- Output head: [G,1] = pooled @ W_out^T + b_out ----------------
__global__ void head_kernel(const float* __restrict__ sums, const float* __restrict__ cnts,
                            const float* __restrict__ Wout, const float* __restrict__ bout,
                            float* __restrict__ out, int G) {
    int g = blockIdx.x * blockDim.x + threadIdx.x;
    if (g >= G) return;
    float inv = 1.0f / fmaxf(cnts[g], 1.0f);
    float acc = bout[0];
#pragma unroll 8
    for (int c = 0; c < H; ++c) acc += sums[g * H + c] * inv * Wout[c];
    out[g] = acc;
}

extern "C" void kernel_launch(void* const* d_in, const int* in_sizes, int n_in,
                              void* d_out, int out_size, void* d_ws, size_t ws_size,
                              hipStream_t stream) {
    const int N = in_sizes[0];
    const int E = in_sizes[1] / 2;
    const int G = out_size;

    const float* x     = (const float*)d_in[0];
    const int*   ei    = (const int*)d_in[1];
    const int*   src   = ei;
    const int*   dst   = ei + E;
    const int*   batch = (const int*)d_in[2];

    // per-layer param base: 3 + (li-1)*7 : Wrel, brel, Wroot, gamma, beta, mean, var
    const float* P[21];
    for (int i = 0; i < 21; ++i) P[i] = (const float*)d_in[3 + i];
    const float* Wout = (const float*)d_in[24];
    const float* bout = (const float*)d_in[25];

    float* bufA = (float*)d_ws;                 // N*H
    float* bufB = bufA + (size_t)N * H;         // N*H
    float* agg1 = bufB + (size_t)N * H;         // N
    float* sums = agg1 + N;                     // G*H
    float* cnts = sums + (size_t)G * H;         // G

    hipMemsetAsync(agg1, 0, (size_t)N * sizeof(float), stream);
    hipMemsetAsync(bufB, 0, (size_t)N * H * sizeof(float), stream);
    hipMemsetAsync(sums, 0, (size_t)(G * H + G) * sizeof(float), stream);

    const int BT = 256;
    // Layer 1
    edge_scatter1<<<(E + BT - 1) / BT, BT, 0, stream>>>(x, src, dst, agg1, E);
    layer1_node<<<(N * H + BT - 1) / BT, BT, 0, stream>>>(
        x, agg1, P[0], P[1], P[2], P[3], P[4], P[5], P[6], bufA, N);

    const int edgeBlocks = (int)(((long long)E * 32 + BT - 1) / BT);
    const int gemmBlocks = (N + 15) / 16;

    // Layer 2: agg into bufB, in-place update bufB (h2)
    edge_scatterH<<<edgeBlocks, BT, 0, stream>>>(bufA, src, dst, bufB, E);
    gemm_bn<true><<<gemmBlocks, BT, 0, stream>>>(
        bufB, bufA, P[7], P[8], P[9], P[10], P[11], P[12], P[13], bufB, N);

    // Layer 3: agg into bufA (zero first; h1 no longer needed), in-place update bufA (h3)
    hipMemsetAsync(bufA, 0, (size_t)N * H * sizeof(float), stream);
    edge_scatterH<<<edgeBlocks, BT, 0, stream>>>(bufB, src, dst, bufA, E);
    gemm_bn<false><<<gemmBlocks, BT, 0, stream>>>(
        bufA, bufB, P[14], P[15], P[16], P[17], P[18], P[19], P[20], bufA, N);

    // Pool + head
    pool_kernel<<<(N * H + BT - 1) / BT, BT, 0, stream>>>(bufA, batch, sums, cnts, N);
    head_kernel<<<(G + 63) / 64, 64, 0, stream>>>(sums, cnts, Wout, bout, (float*)d_out, G);
}